// KnowledgeBaseModule_88742614270240
// MI455X (gfx1250) — compile-verified
//
#include <hip/hip_runtime.h>
#include <hip/hip_bf16.h>
#include <math.h>

typedef float v2f __attribute__((ext_vector_type(2)));
typedef float v8f __attribute__((ext_vector_type(8)));
typedef int   aint4 __attribute__((vector_size(16)));   // builtin's pointee type

#define HDIM 1024
#define H2   2048
#define BM   16          // rows per block tile
#define TC   256         // intermediate (2H) chunk width
#define NCH  (H2 / TC)   // 8 chunks
#define XPAD 4           // LDS row padding (stride % 64 == 4 -> conflict-free)
#define TPAD 4
#define W1STR 36         // stage-1 slab row stride (32 cols + 4 pad)
#define W2STR 132        // stage-2 slab row stride (128 cols + 4 pad)
#define WSLAB 1152       // floats per slab buffer = max(32*36, 8*132)
#define NENT 1000
#define TOPK 5
#define BB   8
#define SEQ  2048

// ---------------- async global->LDS copy path (gfx1250 ASYNCcnt) ----------
#if __has_builtin(__builtin_amdgcn_global_load_async_to_lds_b128)
#define HAVE_ASYNC 1
#define ASYNC_COPY16(g, l) \
    __builtin_amdgcn_global_load_async_to_lds_b128((aint4*)(g), (aint4*)(l), 0, 0)
#define ASYNC_WAIT8() asm volatile("s_wait_asynccnt 0x8" ::: "memory")
#define ASYNC_WAIT0() asm volatile("s_wait_asynccnt 0x0" ::: "memory")
#else
#define HAVE_ASYNC 0
#define ASYNC_COPY16(g, l) (*(float4*)(l) = *(const float4*)(g))
#define ASYNC_WAIT8() ((void)0)
#define ASYNC_WAIT0() ((void)0)
#endif

__device__ __forceinline__ float gelu_f(float x) {
    return 0.5f * x * (1.0f + erff(x * 0.70710678118654752f));
}

// per-wave: stage 32k x 32col slab of W1 (256 x b128, 8 per lane)
__device__ __forceinline__ void issue_w1_slice(const float* __restrict__ W1,
                                               int k0, int ccol0,
                                               float* slab, int lane)
{
#pragma unroll
    for (int i = 0; i < 8; ++i) {
        const int e  = lane * 8 + i;   // 0..255
        const int r  = e >> 3;         // 32 rows
        const int c4 = e & 7;          // 8 x float4 per row
        ASYNC_COPY16(W1 + (long)(k0 + r) * H2 + ccol0 + c4 * 4,
                     slab + r * W1STR + c4 * 4);
    }
}

// per-wave: stage 8k x 128col slab of W2 (256 x b128, 8 per lane)
__device__ __forceinline__ void issue_w2_slice(const float* __restrict__ W2,
                                               int k0, int ocol0,
                                               float* slab, int lane)
{
#pragma unroll
    for (int i = 0; i < 8; ++i) {
        const int e  = lane * 8 + i;   // 0..255
        const int r  = e >> 5;         // 8 rows
        const int c4 = e & 31;         // 32 x float4 per row
        ASYNC_COPY16(W2 + (long)(k0 + r) * HDIM + ocol0 + c4 * 4,
                     slab + r * W2STR + c4 * 4);
    }
}

// ---------------------------------------------------------------------------
// Fused entity encoder: LN(gelu(X@W1 + b1)@W2 + b2) for X = [16384,1024]
// One block = 16 rows, 8 waves. WMMA f32 16x16x4 + async-LDS weight streaming.
// ---------------------------------------------------------------------------
__global__ __launch_bounds__(256)
void entity_encoder_kernel(const float* __restrict__ X,    // [16384][1024]
                           const float* __restrict__ W1,   // [1024][2048]
                           const float* __restrict__ B1,   // [2048]
                           const float* __restrict__ W2,   // [2048][1024]
                           const float* __restrict__ B2,   // [1024]
                           const float* __restrict__ G,
                           const float* __restrict__ Beta,
                           float* __restrict__ OutG)       // [16384][1024]
{
    __shared__ __align__(16) float Xs[BM][HDIM + XPAD];      // ~65.8 KB, reused for pre-LN out
    __shared__ __align__(16) float Ts[BM][TC + TPAD];        // ~16.6 KB
    __shared__ __align__(16) float Wst[8][2][WSLAB];         // ~73.7 KB per-wave double buffers

    const int tid  = threadIdx.x;
    const int lane = tid & 31;
    const int wv   = tid >> 5;        // wave 0..7
    const int kh   = lane >> 4;       // lane half (0/1)
    const int l16  = lane & 15;
    const long rowBase = (long)blockIdx.x * BM;

    // ---- stage 0: async-load X tile into LDS ----
    {
        const int nq = HDIM / 4;                 // 256 float4s per row
        for (int i = tid; i < BM * nq; i += 256) {
            const int r = i >> 8, q = i & 255;
            ASYNC_COPY16(X + (rowBase + r) * HDIM + q * 4, &Xs[r][q * 4]);
        }
        ASYNC_WAIT0();
    }
    __syncthreads();

    const v8f vzero = {0.f, 0.f, 0.f, 0.f, 0.f, 0.f, 0.f, 0.f};
    v8f acc2[8];
#pragma unroll
    for (int t = 0; t < 8; ++t) acc2[t] = vzero;

    float* const slab0 = &Wst[wv][0][0];
    float* const slab1 = &Wst[wv][1][0];

    for (int c = 0; c < NCH; ++c) {
        // =====================================================================
        // stage 1: Ts[16][256] = gelu(Xs @ W1[:, c*TC ..] + b1)
        // wave owns 32 chunk columns; W1 streamed per-wave, double-buffered.
        // =====================================================================
        v8f acc1[2];
        acc1[0] = vzero; acc1[1] = vzero;
        const int ccol0 = c * TC + wv * 32;

        issue_w1_slice(W1, 0,  ccol0, slab0, lane);
        issue_w1_slice(W1, 32, ccol0, slab1, lane);

        for (int s = 0; s < 32; ++s) {
            if (s < 31) { ASYNC_WAIT8(); } else { ASYNC_WAIT0(); }
            float* myW = (s & 1) ? slab1 : slab0;
#pragma unroll
            for (int k8 = 0; k8 < 8; ++k8) {
                const int rr = k8 * 4 + 2 * kh;          // slab-local k row
                const int kk = s * 32 + rr;              // global k
                v2f a;  a.x = Xs[l16][kk];               a.y = Xs[l16][kk + 1];
                v2f b0; b0.x = myW[rr * W1STR + l16];    b0.y = myW[(rr + 1) * W1STR + l16];
                v2f b1; b1.x = myW[rr * W1STR + 16 + l16];
                        b1.y = myW[(rr + 1) * W1STR + 16 + l16];
                acc1[0] = __builtin_amdgcn_wmma_f32_16x16x4_f32(false, a, false, b0,
                                                                (short)0, acc1[0], false, false);
                acc1[1] = __builtin_amdgcn_wmma_f32_16x16x4_f32(false, a, false, b1,
                                                                (short)0, acc1[1], false, false);
            }
            if (s + 2 < 32) issue_w1_slice(W1, (s + 2) * 32, ccol0,
                                           (s & 1) ? slab1 : slab0, lane);
        }

        // bias + gelu -> Ts
#pragma unroll
        for (int t = 0; t < 2; ++t) {
            const int gcol = ccol0 + t * 16 + l16;
            const float bias = B1[gcol];
            const int colc = wv * 32 + t * 16 + l16;
#pragma unroll
            for (int r = 0; r < 8; ++r) {
                Ts[r + 8 * kh][colc] = gelu_f(acc1[t][r] + bias);
            }
        }

        // =====================================================================
        // stage 2: acc2 += Ts @ W2[c*TC .., :]   (wave owns 128 out cols)
        // =====================================================================
        const int ocol0 = wv * 128;
        issue_w2_slice(W2, c * TC + 0, ocol0, slab0, lane);
        issue_w2_slice(W2, c * TC + 8, ocol0, slab1, lane);
        __syncthreads();            // Ts visible to all waves

        for (int s = 0; s < 32; ++s) {
            if (s < 31) { ASYNC_WAIT8(); } else { ASYNC_WAIT0(); }
            float* myW = (s & 1) ? slab1 : slab0;
#pragma unroll
            for (int k2 = 0; k2 < 2; ++k2) {
                const int rr = k2 * 4 + 2 * kh;          // slab-local k row
                const int kk = s * 8 + rr;               // chunk-local k
                v2f a; a.x = Ts[l16][kk]; a.y = Ts[l16][kk + 1];
#pragma unroll
                for (int t = 0; t < 8; ++t) {
                    v2f b; b.x = myW[rr * W2STR + t * 16 + l16];
                           b.y = myW[(rr + 1) * W2STR + t * 16 + l16];
                    acc2[t] = __builtin_amdgcn_wmma_f32_16x16x4_f32(false, a, false, b,
                                                                    (short)0, acc2[t], false, false);
                }
            }
            if (s + 2 < 32) issue_w2_slice(W2, c * TC + (s + 2) * 8, ocol0,
                                           (s & 1) ? slab1 : slab0, lane);
        }
        __syncthreads();            // Ts free for next chunk
    }

    // ---- bias + stash pre-LN tile into Xs (reuse) ----
#pragma unroll
    for (int t = 0; t < 8; ++t) {
        const int ocol = wv * 128 + t * 16 + l16;
        const float bias = B2[ocol];
#pragma unroll
        for (int r = 0; r < 8; ++r) {
            Xs[r + 8 * kh][ocol] = acc2[t][r] + bias;
        }
    }
    __syncthreads();

    // ---- fused LayerNorm: wave wv handles rows 2wv, 2wv+1 ----
    for (int rr = 0; rr < 2; ++rr) {
        const int row = wv * 2 + rr;
        float s = 0.f, q = 0.f;
        for (int kc = lane; kc < HDIM; kc += 32) {
            float v = Xs[row][kc];
            s += v; q += v * v;
        }
#pragma unroll
        for (int off = 16; off > 0; off >>= 1) {
            s += __shfl_down(s, off, 32);
            q += __shfl_down(q, off, 32);
        }
        s = __shfl(s, 0, 32);
        q = __shfl(q, 0, 32);
        const float mu  = s * (1.0f / HDIM);
        const float var = q * (1.0f / HDIM) - mu * mu;
        const float rs  = rsqrtf(var + 1e-5f);
        const long gRow = rowBase + row;
        for (int kc = lane; kc < HDIM; kc += 32) {
            float v = (Xs[row][kc] - mu) * rs;
            OutG[gRow * HDIM + kc] = v * G[kc] + Beta[kc];
        }
    }
}

// ---------------------------------------------------------------------------
// pooled[b][h] = mean over S of hidden[b][s][h]
// ---------------------------------------------------------------------------
__global__ __launch_bounds__(256)
void pool_kernel(const float* __restrict__ X, float* __restrict__ pooled)
{
    int idx = blockIdx.x * 256 + threadIdx.x;     // 8192 total
    int b = idx >> 10, h = idx & (HDIM - 1);
    const float* p = X + ((long)b * SEQ) * HDIM + h;
    float s = 0.f;
    for (int i = 0; i < SEQ; ++i) s += p[(long)i * HDIM];
    pooled[idx] = s * (1.0f / SEQ);
}

// ---------------------------------------------------------------------------
// generic dense: out[b][j] = act(sum_k in[b][k]*W[k][j] + bias[j])
// act: 0=none, 1=gelu, 2=sigmoid
// ---------------------------------------------------------------------------
__global__ __launch_bounds__(256)
void dense_kernel(const float* __restrict__ in, const float* __restrict__ W,
                  const float* __restrict__ bias, float* __restrict__ out,
                  int K, int N, int act)
{
    int j = blockIdx.x * 256 + threadIdx.x;
    if (j >= N) return;
    int b = blockIdx.y;
    const float* x = in + (long)b * K;
    float s = bias[j];
    for (int k = 0; k < K; ++k) s += x[k] * W[(long)k * N + j];
    if (act == 1)      s = gelu_f(s);
    else if (act == 2) s = 1.0f / (1.0f + expf(-s));
    out[(long)b * N + j] = s;
}

// ---------------------------------------------------------------------------
// row LayerNorm over N, one block per batch row
// ---------------------------------------------------------------------------
__global__ __launch_bounds__(256)
void ln_kernel(const float* __restrict__ in, const float* __restrict__ g,
               const float* __restrict__ beta, float* __restrict__ out, int N)
{
    __shared__ float ss[256], sq[256];
    int b = blockIdx.x, tid = threadIdx.x;
    const float* x = in + (long)b * N;
    float s = 0.f, q = 0.f;
    for (int k = tid; k < N; k += 256) { float v = x[k]; s += v; q += v * v; }
    ss[tid] = s; sq[tid] = q; __syncthreads();
    for (int st = 128; st > 0; st >>= 1) {
        if (tid < st) { ss[tid] += ss[tid + st]; sq[tid] += sq[tid + st]; }
        __syncthreads();
    }
    float mu  = ss[0] / N;
    float var = sq[0] / N - mu * mu;
    float rs  = rsqrtf(var + 1e-5f);
    for (int k = tid; k < N; k += 256)
        out[(long)b * N + k] = (x[k] - mu) * rs * g[k] + beta[k];
}

// ---------------------------------------------------------------------------
// sims[b][e] = dot(relation_features[b], ent[e]); one wave per entity
// ---------------------------------------------------------------------------
__global__ __launch_bounds__(256)
void sims_kernel(const float* __restrict__ rf, const float* __restrict__ ent,
                 float* __restrict__ sims)
{
    int wv = threadIdx.x >> 5, lane = threadIdx.x & 31;
    int e = blockIdx.x * 8 + wv;
    int b = blockIdx.y;
    if (e >= NENT) return;
    const float* er = ent + (long)e * HDIM;
    const float* xr = rf + (long)b * HDIM;
    float s = 0.f;
    for (int k = lane; k < HDIM; k += 32) s += er[k] * xr[k];
#pragma unroll
    for (int off = 16; off > 0; off >>= 1) s += __shfl_down(s, off, 32);
    if (lane == 0) sims[b * NENT + e] = s;
}

// ---------------------------------------------------------------------------
// top-5 (lowest-index tie break, jax semantics) + gather + mean entity vec
// ---------------------------------------------------------------------------
__global__ __launch_bounds__(256)
void topk_kernel(const float* __restrict__ sims, const float* __restrict__ ent,
                 float* __restrict__ out_idx_f, float* __restrict__ retrieved,
                 float* __restrict__ entity_vec)
{
    __shared__ float sval[256];
    __shared__ int   sidx[256];
    __shared__ int   chosen[TOPK];
    int b = blockIdx.x, tid = threadIdx.x;

    for (int t = 0; t < TOPK; ++t) {
        float best = -3.4e38f; int bi = 0x7fffffff;
        for (int e = tid; e < NENT; e += 256) {
            bool skip = false;
            for (int q = 0; q < t; ++q) if (chosen[q] == e) skip = true;
            if (skip) continue;
            float v = sims[b * NENT + e];
            if (v > best || (v == best && e < bi)) { best = v; bi = e; }
        }
        sval[tid] = best; sidx[tid] = bi;
        __syncthreads();
        for (int st = 128; st > 0; st >>= 1) {
            if (tid < st) {
                if (sval[tid + st] > sval[tid] ||
                    (sval[tid + st] == sval[tid] && sidx[tid + st] < sidx[tid])) {
                    sval[tid] = sval[tid + st]; sidx[tid] = sidx[tid + st];
                }
            }
            __syncthreads();
        }
        if (tid == 0) chosen[t] = sidx[0];
        __syncthreads();
    }

    if (tid < TOPK) out_idx_f[b * TOPK + tid] = (float)chosen[tid];

    for (int h = tid; h < HDIM; h += 256) {
        float acc = 0.f;
        for (int t = 0; t < TOPK; ++t) {
            float v = ent[(long)chosen[t] * HDIM + h];
            retrieved[((long)b * TOPK + t) * HDIM + h] = v;
            acc += v;
        }
        entity_vec[b * HDIM + h] = acc * 0.2f;
    }
}

// ---------------------------------------------------------------------------
// reasoning_in = [entity_vec | relation_features | pooled]  -> [8][3072]
// ---------------------------------------------------------------------------
__global__ __launch_bounds__(256)
void concat_kernel(const float* __restrict__ ev, const float* __restrict__ rf,
                   const float* __restrict__ pooled, float* __restrict__ rin)
{
    int idx = blockIdx.x * 256 + threadIdx.x;    // 8*3072
    int b = idx / (3 * HDIM), c = idx % (3 * HDIM);
    float v;
    if (c < HDIM)           v = ev[b * HDIM + c];
    else if (c < 2 * HDIM)  v = rf[b * HDIM + (c - HDIM)];
    else                    v = pooled[b * HDIM + (c - 2 * HDIM)];
    rin[idx] = v;
}

// ---------------------------------------------------------------------------
extern "C" void kernel_launch(void* const* d_in, const int* in_sizes, int n_in,
                              void* d_out, int out_size, void* d_ws, size_t ws_size,
                              hipStream_t stream)
{
    const float* hs      = (const float*)d_in[0];
    const float* ent     = (const float*)d_in[1];
    const float* ee_w1   = (const float*)d_in[2];
    const float* ee_b1   = (const float*)d_in[3];
    const float* ee_w2   = (const float*)d_in[4];
    const float* ee_b2   = (const float*)d_in[5];
    const float* ee_g    = (const float*)d_in[6];
    const float* ee_beta = (const float*)d_in[7];
    const float* re_w1   = (const float*)d_in[8];
    const float* re_b1   = (const float*)d_in[9];
    const float* re_w2   = (const float*)d_in[10];
    const float* re_b2   = (const float*)d_in[11];
    const float* re_g    = (const float*)d_in[12];
    const float* re_beta = (const float*)d_in[13];
    const float* rn_w1   = (const float*)d_in[14];
    const float* rn_b1   = (const float*)d_in[15];
    const float* rn_w2   = (const float*)d_in[16];
    const float* rn_b2   = (const float*)d_in[17];
    const float* rn_g    = (const float*)d_in[18];
    const float* rn_beta = (const float*)d_in[19];
    const float* vn_w1   = (const float*)d_in[20];
    const float* vn_b1   = (const float*)d_in[21];
    const float* vn_w2   = (const float*)d_in[22];
    const float* vn_b2   = (const float*)d_in[23];

    float* out = (float*)d_out;
    // output layout (flat, reference return order)
    float* out_ef  = out;                        // [8,2048,1024] = 16777216
    float* out_rf  = out + 16777216;             // [8,1024]
    float* out_ret = out + 16785408;             // [8,5,1024]
    float* out_sim = out + 16826368;             // [8,1000]
    float* out_idx = out + 16834368;             // [8,5] (as float)
    float* out_ro  = out + 16834408;             // [8,1024]
    float* out_vs  = out + 16842600;             // [8,1]

    float* ws = (float*)d_ws;
    float* ws_pooled = ws;               // 8192
    float* ws_t1     = ws + 8192;        // 8192
    float* ws_t2     = ws + 16384;       // 8192
    float* ws_ev     = ws + 24576;       // 8192
    float* ws_rin    = ws + 32768;       // 24576
    float* ws_rn1    = ws + 57344;       // 16384
    float* ws_rn2    = ws + 73728;       // 8192
    float* ws_vn1    = ws + 81920;       // 4096

    // 1) entity encoder (dominant WMMA kernel): 16384/16 = 1024 blocks
    entity_encoder_kernel<<<dim3(16384 / BM), 256, 0, stream>>>(
        hs, ee_w1, ee_b1, ee_w2, ee_b2, ee_g, ee_beta, out_ef);

    // 2) mean pool over sequence
    pool_kernel<<<dim3(BB * HDIM / 256), 256, 0, stream>>>(hs, ws_pooled);

    // 3) relation encoder
    dense_kernel<<<dim3(HDIM / 256, BB), 256, 0, stream>>>(ws_pooled, re_w1, re_b1, ws_t1, HDIM, HDIM, 1);
    dense_kernel<<<dim3(HDIM / 256, BB), 256, 0, stream>>>(ws_t1, re_w2, re_b2, ws_t2, HDIM, HDIM, 0);
    ln_kernel<<<dim3(BB), 256, 0, stream>>>(ws_t2, re_g, re_beta, out_rf, HDIM);

    // 4) entity similarities
    sims_kernel<<<dim3((NENT + 7) / 8, BB), 256, 0, stream>>>(out_rf, ent, out_sim);

    // 5) top-k + gather + mean entity vector
    topk_kernel<<<dim3(BB), 256, 0, stream>>>(out_sim, ent, out_idx, out_ret, ws_ev);

    // 6) reasoning input concat
    concat_kernel<<<dim3(BB * 3 * HDIM / 256), 256, 0, stream>>>(ws_ev, out_rf, ws_pooled, ws_rin);

    // 7) reasoning network
    dense_kernel<<<dim3(H2 / 256, BB), 256, 0, stream>>>(ws_rin, rn_w1, rn_b1, ws_rn1, 3 * HDIM, H2, 1);
    dense_kernel<<<dim3(HDIM / 256, BB), 256, 0, stream>>>(ws_rn1, rn_w2, rn_b2, ws_rn2, H2, HDIM, 0);
    ln_kernel<<<dim3(BB), 256, 0, stream>>>(ws_rn2, rn_g, rn_beta, out_ro, HDIM);

    // 8) validation network
    dense_kernel<<<dim3((HDIM / 2) / 256, BB), 256, 0, stream>>>(out_ro, vn_w1, vn_b1, ws_vn1, HDIM, HDIM / 2, 1);
    dense_kernel<<<dim3(1, BB), 256, 0, stream>>>(ws_vn1, vn_w2, vn_b2, out_vs, HDIM / 2, 1, 2);
}